// OBBIoULoss_76055280877918
// MI455X (gfx1250) — compile-verified
//
#include <hip/hip_runtime.h>
#include <math.h>

#define EPSF      1e-8f
#define MODE_EPSF 1e-6f

typedef float v2f __attribute__((ext_vector_type(2)));
typedef float v8f __attribute__((ext_vector_type(8)));

__device__ __forceinline__ float wave_reduce_add(float v) {
    // wave32 on gfx1250
    #pragma unroll
    for (int off = 16; off > 0; off >>= 1)
        v += __shfl_down(v, off, 32);
    return v;
}

// Monotone surrogate for atan2f(dy, dx): maps direction to (-2, 2],
// strictly increasing with the true angle. Used only for ordering.
__device__ __forceinline__ float pseudo_angle(float dy, float dx) {
    float ax = fabsf(dx) + fabsf(dy);
    float r  = (ax == 0.0f) ? 0.0f : (dy / ax);        // in [-1, 1]
    // dx >= 0            : angle in [-pi/2, pi/2]  -> r            in [-1, 1]
    // dx <  0, dy >= 0   : angle in (pi/2, pi]     -> 2 - r        in (1, 2]
    // dx <  0, dy <  0   : angle in (-pi, -pi/2)   -> -2 - r       in (-2, -1)
    return (dx >= 0.0f) ? r : ((dy >= 0.0f) ? (2.0f - r) : (-2.0f - r));
}

// -------------------------------------------------------------------------
// Kernel 1: one thread per box pair -> full rotated CIoU loss, block-reduce
// into per-block partial sums. All candidate-vertex arrays are constant-index
// fully-unrolled so they live in VGPRs (no scratch).
// -------------------------------------------------------------------------
__global__ void __launch_bounds__(256)
obb_ciou_loss_kernel(const float* __restrict__ pred,
                     const float* __restrict__ target,
                     const float* __restrict__ weight,
                     float* __restrict__ partial, int N)
{
    const int idx = blockIdx.x * blockDim.x + threadIdx.x;
    float loss = 0.0f;

    if (idx < N) {
        const float* p = pred   + (size_t)idx * 5;
        const float* t = target + (size_t)idx * 5;
        const float pcx = p[0], pcy = p[1], pw = p[2], ph = p[3], pa = p[4];
        const float tcx = t[0], tcy = t[1], tw = t[2], th = t[3], ta = t[4];

        // ---- box2corners ----
        const float hx[4] = { 0.5f, -0.5f, -0.5f,  0.5f};
        const float hy[4] = {-0.5f, -0.5f,  0.5f,  0.5f};
        float s1 = sinf(pa), co1 = cosf(pa);
        float s2 = sinf(ta), co2 = cosf(ta);
        float c1x[4], c1y[4], c2x[4], c2y[4];
        #pragma unroll
        for (int i = 0; i < 4; ++i) {
            float xx = hx[i] * pw, yy = hy[i] * ph;
            c1x[i] = xx * co1 - yy * s1 + pcx;
            c1y[i] = xx * s1  + yy * co1 + pcy;
            xx = hx[i] * tw; yy = hy[i] * th;
            c2x[i] = xx * co2 - yy * s2 + tcx;
            c2y[i] = xx * s2  + yy * co2 + tcy;
        }

        // ---- candidate vertices: 8 corners + 16 edge intersections ----
        float vx[24], vy[24];
        bool  vm[24];
        #pragma unroll
        for (int i = 0; i < 4; ++i) { vx[i]     = c1x[i]; vy[i]     = c1y[i]; }
        #pragma unroll
        for (int i = 0; i < 4; ++i) { vx[4 + i] = c2x[i]; vy[4 + i] = c2y[i]; }

        // box_in_box(corners1, corners2): corners of box1 inside box2
        {
            float abx = c2x[1] - c2x[0], aby = c2y[1] - c2y[0];
            float adx = c2x[3] - c2x[0], ady = c2y[3] - c2y[0];
            float ab2 = abx * abx + aby * aby;
            float ad2 = adx * adx + ady * ady;
            #pragma unroll
            for (int k = 0; k < 4; ++k) {
                float amx = c1x[k] - c2x[0], amy = c1y[k] - c2y[0];
                float pab = (abx * amx + aby * amy) / ab2;
                float pad = (adx * amx + ady * amy) / ad2;
                vm[k] = (pab > -1e-6f) && (pab < 1.000001f) &&
                        (pad > -1e-6f) && (pad < 1.000001f);
            }
        }
        // box_in_box(corners2, corners1): corners of box2 inside box1
        {
            float abx = c1x[1] - c1x[0], aby = c1y[1] - c1y[0];
            float adx = c1x[3] - c1x[0], ady = c1y[3] - c1y[0];
            float ab2 = abx * abx + aby * aby;
            float ad2 = adx * adx + ady * ady;
            #pragma unroll
            for (int k = 0; k < 4; ++k) {
                float amx = c2x[k] - c1x[0], amy = c2y[k] - c1y[0];
                float pab = (abx * amx + aby * amy) / ab2;
                float pad = (adx * amx + ady * amy) / ad2;
                vm[4 + k] = (pab > -1e-6f) && (pab < 1.000001f) &&
                            (pad > -1e-6f) && (pad < 1.000001f);
            }
        }
        // 16 segment-segment intersections (edges of box1 x edges of box2)
        #pragma unroll
        for (int i = 0; i < 4; ++i) {
            const float x1 = c1x[i], y1 = c1y[i];
            const float x2 = c1x[(i + 1) & 3], y2 = c1y[(i + 1) & 3];
            #pragma unroll
            for (int j = 0; j < 4; ++j) {
                const float x3 = c2x[j], y3 = c2y[j];
                const float x4 = c2x[(j + 1) & 3], y4 = c2y[(j + 1) & 3];
                float num  = (y4 - y3) * (x2 - x1) - (x4 - x3) * (y2 - y1);
                float dent = (x4 - x3) * (y1 - y3) - (y4 - y3) * (x1 - x3);
                float tt   = dent / (num + EPSF);
                bool  mt   = (num != 0.0f) && (tt > 0.0f) && (tt < 1.0f);
                float denu = (x2 - x1) * (y1 - y3) - (y2 - y1) * (x1 - x3);
                float uu   = -denu / (num + EPSF);
                bool  mu   = (num != 0.0f) && (uu > 0.0f) && (uu < 1.0f);
                bool  m    = mt && mu;
                const int k = 8 + i * 4 + j;
                vm[k] = m;
                vx[k] = m ? (x1 + tt * (x2 - x1)) : 0.0f;
                vy[k] = m ? (y1 + tt * (y2 - y1)) : 0.0f;
            }
        }

        // ---- masked centroid ----
        float nvf = 0.0f, sx = 0.0f, sy = 0.0f;
        #pragma unroll
        for (int k = 0; k < 24; ++k) {
            float m = vm[k] ? 1.0f : 0.0f;
            nvf += m; sx += vx[k] * m; sy += vy[k] * m;
        }
        const float inv_nv = 1.0f / fmaxf(nvf, 1.0f);
        const float cxm = sx * inv_nv, cym = sy * inv_nv;
        const int nvalid = (int)nvf;

        // ---- sort keys (monotone pseudo-angle; invalid -> 1e7) ----
        float key[24], rx[24], ry[24];
        #pragma unroll
        for (int k = 0; k < 24; ++k) {
            float dx = vx[k] - cxm, dy = vy[k] - cym;
            rx[k] = dx; ry[k] = dy;
            key[k] = vm[k] ? pseudo_angle(dy, dx) : 10000000.0f;
        }

        // ---- Batcher odd-even mergesort, n=24 (~130 comparators) ----
        // Iterative arbitrary-n form; all indices constant after unroll.
        {
            const int n = 24;
            #pragma unroll
            for (int pp = 1; pp < n; pp <<= 1) {
                #pragma unroll
                for (int kk = pp; kk >= 1; kk >>= 1) {
                    #pragma unroll
                    for (int j = kk % pp; j + kk < n; j += 2 * kk) {
                        #pragma unroll
                        for (int i = 0; i < kk; ++i) {
                            const int lo = i + j, hi = i + j + kk;
                            if (hi < n && (lo / (pp * 2)) == (hi / (pp * 2))) {
                                float ka = key[lo], kb = key[hi];
                                bool sw = ka > kb;
                                float xa = rx[lo], xb = rx[hi];
                                float ya = ry[lo], yb = ry[hi];
                                key[lo] = sw ? kb : ka;  key[hi] = sw ? ka : kb;
                                rx[lo]  = sw ? xb : xa;  rx[hi]  = sw ? xa : xb;
                                ry[lo]  = sw ? yb : ya;  ry[hi]  = sw ? ya : yb;
                            }
                        }
                    }
                }
            }
        }

        // ---- shoelace: entries >= nvalid collapse to first vertex ----
        const float fx = rx[0], fy = ry[0];
        float qx[24], qy[24];
        #pragma unroll
        for (int k = 0; k < 24; ++k) {
            qx[k] = (k < nvalid) ? rx[k] : fx;
            qy[k] = (k < nvalid) ? ry[k] : fy;
        }
        float cr = 0.0f;
        #pragma unroll
        for (int k = 0; k < 24; ++k) {
            const int kn = (k + 1) % 24;
            cr += qx[k] * qy[kn] - qy[k] * qx[kn];
        }
        const float inter = fabsf(cr) * 0.5f;

        // ---- CIoU tail ----
        const float area1 = pw * ph, area2 = tw * th;
        float iou = inter / (area1 + area2 - inter);
        iou = fminf(fmaxf(iou, 0.0f), 1.0f);

        const float ca1 = fabsf(co1), sa1 = fabsf(s1);
        const float ca2 = fabsf(co2), sa2 = fabsf(s2);
        const float dw1 = (pw * ca1 + ph * sa1) * 0.5f, dh1 = (pw * sa1 + ph * ca1) * 0.5f;
        const float dw2 = (tw * ca2 + th * sa2) * 0.5f, dh2 = (tw * sa2 + th * ca2) * 0.5f;
        float enw = fmaxf(pcx + dw1, tcx + dw2) - fminf(pcx - dw1, tcx - dw2);
        float enh = fmaxf(pcy + dh1, tcy + dh2) - fminf(pcy - dh1, tcy - dh2);
        enw = fmaxf(enw, 0.0f); enh = fmaxf(enh, 0.0f);
        const float c2v  = enw * enw + enh * enh + MODE_EPSF;
        const float rho2 = (tcx - pcx) * (tcx - pcx) + (tcy - pcy) * (tcy - pcy);
        const float FACTOR = 0.4052847345693511f;  // 4 / pi^2
        const float dv = atanf(tw / (th + MODE_EPSF)) - atanf(pw / (ph + MODE_EPSF));
        const float v  = FACTOR * dv * dv;
        const float alpha = (iou > 0.5f) ? (v / (1.0f - iou + v + MODE_EPSF)) : 0.0f;
        const float ciou  = iou - (fminf(fmaxf(rho2 / c2v, 0.0f), 1.0f) + alpha * v);
        loss = (1.0f - ciou) * weight[idx];
    }

    // ---- block reduction -> per-block partial sum ----
    __shared__ float wsum[8];
    float wr = wave_reduce_add(loss);
    const int lane = threadIdx.x & 31;
    const int wave = threadIdx.x >> 5;
    if (lane == 0) wsum[wave] = wr;
    __syncthreads();
    if (wave == 0) {
        float a = (lane < 8) ? wsum[lane] : 0.0f;
        a = wave_reduce_add(a);
        if (lane == 0) partial[blockIdx.x] = a;
    }
}

// -------------------------------------------------------------------------
// Kernel 2: single-block finish. Strided sum of block partials, shuffle
// reduce per wave, then a V_WMMA_F32_16X16X4_F32 finish: A holds the 8 wave
// sums (rows 0..7, k=0), B = ones(4x16), so D's accumulator VGPRs on lane 0
// hold the row sums -> exact f32 total. Writes mean to out[0].
// -------------------------------------------------------------------------
__global__ void __launch_bounds__(256)
final_reduce_kernel(const float* __restrict__ partial, int nPart,
                    float* __restrict__ out, float invN)
{
    __shared__ float wsum[8];
    float acc = 0.0f;
    for (int i = threadIdx.x; i < nPart; i += 256) acc += partial[i];
    acc = wave_reduce_add(acc);
    const int lane = threadIdx.x & 31;
    const int wave = threadIdx.x >> 5;
    if (lane == 0) wsum[wave] = acc;
    __syncthreads();

    if (wave == 0) {  // wave-uniform branch: EXEC all ones inside (WMMA-legal)
        v2f a;
        a.x = (lane < 8) ? wsum[lane] : 0.0f;  // A: rows 0..7 at K=0
        a.y = 0.0f;                            // K=1 (lanes 0-15) / K=3 (16-31)
        v2f b; b.x = 1.0f; b.y = 1.0f;         // B = ones(4x16)
        v8f c = {};
        // D[m][n] = sum_k A[m][k] -> lane 0, VGPRs 0..7 = wave sums 0..7
        c = __builtin_amdgcn_wmma_f32_16x16x4_f32(
                /*neg_a=*/false, a, /*neg_b=*/false, b,
                /*c_mod=*/(short)0, c, /*reuse_a=*/false, /*reuse_b=*/false);
        float tot = c[0] + c[1] + c[2] + c[3] + c[4] + c[5] + c[6] + c[7];
        if (lane == 0) out[0] = tot * invN;
    }
}

// -------------------------------------------------------------------------
extern "C" void kernel_launch(void* const* d_in, const int* in_sizes, int n_in,
                              void* d_out, int out_size, void* d_ws, size_t ws_size,
                              hipStream_t stream) {
    const float* pred   = (const float*)d_in[0];
    const float* target = (const float*)d_in[1];
    const float* weight = (const float*)d_in[2];
    const int N = in_sizes[2];                 // weight has N elements
    const int nBlocks = (N + 255) / 256;
    float* partial = (float*)d_ws;             // nBlocks floats, rewritten fully each call

    obb_ciou_loss_kernel<<<nBlocks, 256, 0, stream>>>(pred, target, weight, partial, N);
    final_reduce_kernel<<<1, 256, 0, stream>>>(partial, nBlocks, (float*)d_out,
                                               1.0f / (float)N);
}